// HeteroRGCNLayer_21492016349636
// MI455X (gfx1250) — compile-verified
//
#include <hip/hip_runtime.h>

#define NNODES 50000
#define NEDGES 200000
#define NETYPE 3
#define DIN    256
#define DOUT   256

typedef __attribute__((ext_vector_type(2))) float v2f;
typedef __attribute__((ext_vector_type(8))) float v8f;

__device__ __forceinline__ void atomic_add_f32(float* p, float v) {
    // non-returning fp32 atomic -> global_atomic_add_f32 (STOREcnt path)
    __hip_atomic_fetch_add(p, v, __ATOMIC_RELAXED, __HIP_MEMORY_SCOPE_AGENT);
}

// ---------------------------------------------------------------- counts ----
__global__ __launch_bounds__(256) void zero_counts_kernel(float* __restrict__ cnt) {
    int i = blockIdx.x * 256 + threadIdx.x;
    if (i < NETYPE * NNODES) cnt[i] = 0.0f;
}

__global__ __launch_bounds__(256) void degree_kernel(const int* __restrict__ dst,
                                                     float* __restrict__ cnt) {
    int i = blockIdx.x * 256 + threadIdx.x;
    if (i < NETYPE * NEDGES) {
        int e = i / NEDGES;
        atomic_add_f32(&cnt[(size_t)e * NNODES + dst[i]], 1.0f);
    }
}

// ------------------------------------------------------------ fused GEMM ----
// feat @ [W0 | W1 | W2 | W3] via V_WMMA_F32_16X16X4_F32.
// grid = (50000/16 M-tiles, 4); block = 128 (4 waves).
// Each wave owns a 16x64 output tile (4 accumulators): one A fragment is
// reused by 4 WMMAs per K-step, and the 4 accumulator chains are independent
// so the matrix pipe stays busy. 64-col wave tiles never straddle a weight
// matrix (256 cols each), so mat selection is wave-uniform.
__global__ __launch_bounds__(128) void gemm_wmma_kernel(
    const float* __restrict__ feat,
    const float* __restrict__ W0, const float* __restrict__ b0,
    const float* __restrict__ W,  const float* __restrict__ b,
    float* __restrict__ out, float* __restrict__ wh)
{
    const int wave  = threadIdx.x >> 5;
    const int lane  = threadIdx.x & 31;
    const int half  = lane >> 4;        // A layout (32-bit 16x4): 0 -> K={0,1}, 1 -> K={2,3}
    const int lrow  = lane & 15;
    const int tile64 = blockIdx.y * 4 + wave;  // 0..15 over 1024 concatenated cols
    const int mat    = tile64 >> 2;            // 0..3 : which weight matrix
    const int col0   = (tile64 & 3) << 6;      // 64-col base within that matrix
    const int mtile  = blockIdx.x;

    const float* Wsel = (mat == 0) ? W0 : (W + (size_t)(mat - 1) * DIN * DOUT);
    const float* bsel = (mat == 0) ? b0 : (b + (size_t)(mat - 1) * DOUT);
    float*       Osel = (mat == 0) ? out : (wh + (size_t)(mat - 1) * NNODES * DOUT);

    const float* arow_ptr = feat + (size_t)(mtile * 16 + lrow) * DIN;
    const int    bcol     = col0 + lrow;   // lane's column within N-tile 0

    v8f acc0 = {0.f,0.f,0.f,0.f,0.f,0.f,0.f,0.f};
    v8f acc1 = acc0, acc2 = acc0, acc3 = acc0;

#pragma unroll 4
    for (int kk = 0; kk < DIN / 4; ++kk) {
        const int kbase = kk * 4 + half * 2;
        // A 16x4 fp32 fragment: lane holds A[M=lrow][kbase], A[M=lrow][kbase+1]
        v2f a = *(const v2f*)(arow_ptr + kbase);
        // B 4x16 fragments for 4 N-tiles: single base, immediate offsets 0/64/128/192
        const float* brow0 = Wsel + (size_t)kbase * DOUT + bcol;
        const float* brow1 = brow0 + DOUT;
        v2f b0v = { brow0[0],   brow1[0]   };
        v2f b1v = { brow0[16],  brow1[16]  };
        v2f b2v = { brow0[32],  brow1[32]  };
        v2f b3v = { brow0[48],  brow1[48]  };
        acc0 = __builtin_amdgcn_wmma_f32_16x16x4_f32(false, a, false, b0v, (short)0, acc0, false, false);
        acc1 = __builtin_amdgcn_wmma_f32_16x16x4_f32(false, a, false, b1v, (short)0, acc1, false, false);
        acc2 = __builtin_amdgcn_wmma_f32_16x16x4_f32(false, a, false, b2v, (short)0, acc2, false, false);
        acc3 = __builtin_amdgcn_wmma_f32_16x16x4_f32(false, a, false, b3v, (short)0, acc3, false, false);
    }

    // D layout: VGPR r -> row (r + half*8); lanes carry N = bcol + t*16
    const float bias0 = bsel[bcol];
    const float bias1 = bsel[bcol + 16];
    const float bias2 = bsel[bcol + 32];
    const float bias3 = bsel[bcol + 48];
    float* obase = Osel + (size_t)(mtile * 16 + half * 8) * DOUT + bcol;
#pragma unroll
    for (int r = 0; r < 8; ++r) {
        float* orow = obase + (size_t)r * DOUT;
        orow[0]  = acc0[r] + bias0;
        orow[16] = acc1[r] + bias1;
        orow[32] = acc2[r] + bias2;
        orow[48] = acc3[r] + bias3;
    }
}

// --------------------------------------------------------------- scatter ----
// One wave per edge: msg = Wh[e][src] * (w / max(cnt_e[dst],1)), atomically
// accumulated into out (which already holds h0). Lane covers 8 columns.
__global__ __launch_bounds__(256) void scatter_mean_kernel(
    const float* __restrict__ wh,
    const int* __restrict__ src, const int* __restrict__ dst,
    const float* __restrict__ ew, const float* __restrict__ cnt,
    float* __restrict__ out)
{
    const int  wave = threadIdx.x >> 5;
    const int  lane = threadIdx.x & 31;
    const long eid  = (long)blockIdx.x * 8 + wave;
    if (eid >= (long)NETYPE * NEDGES) return;

    const int e = (int)(eid / NEDGES);
    const int i = (int)(eid % NEDGES);
    const int s = src[e * NEDGES + i];
    const int d = dst[e * NEDGES + i];
    const float w = ew[e * NEDGES + i];
    const float c = cnt[(size_t)e * NNODES + d];
    const float scale = w / fmaxf(c, 1.0f);

    const float4* row = (const float4*)(wh + ((size_t)e * NNODES + s) * DOUT) + lane * 2;
    float* orow = out + (size_t)d * DOUT + lane * 8;

    float4 m0 = row[0];
    float4 m1 = row[1];
    atomic_add_f32(orow + 0, m0.x * scale);
    atomic_add_f32(orow + 1, m0.y * scale);
    atomic_add_f32(orow + 2, m0.z * scale);
    atomic_add_f32(orow + 3, m0.w * scale);
    atomic_add_f32(orow + 4, m1.x * scale);
    atomic_add_f32(orow + 5, m1.y * scale);
    atomic_add_f32(orow + 6, m1.z * scale);
    atomic_add_f32(orow + 7, m1.w * scale);
}

// ---------------------------------------------------------------- launch ----
extern "C" void kernel_launch(void* const* d_in, const int* in_sizes, int n_in,
                              void* d_out, int out_size, void* d_ws, size_t ws_size,
                              hipStream_t stream) {
    const float* feat = (const float*)d_in[0];
    const float* W0   = (const float*)d_in[1];
    const float* b0   = (const float*)d_in[2];
    const float* W    = (const float*)d_in[3];
    const float* b    = (const float*)d_in[4];
    const int*   src  = (const int*)d_in[5];
    const int*   dst  = (const int*)d_in[6];
    const float* ew   = (const float*)d_in[7];

    float* out = (float*)d_out;
    float* wh  = (float*)d_ws;                                   // [NE][N][DOUT]
    float* cnt = wh + (size_t)NETYPE * NNODES * DOUT;            // [NE][N]

    // 1) per-etype in-degree counts
    zero_counts_kernel<<<(NETYPE * NNODES + 255) / 256, 256, 0, stream>>>(cnt);
    degree_kernel<<<(NETYPE * NEDGES + 255) / 256, 256, 0, stream>>>(dst, cnt);

    // 2) fused WMMA GEMM: h0 -> out, Wh[e] -> ws
    dim3 grid(NNODES / 16, 4);
    gemm_wmma_kernel<<<grid, 128, 0, stream>>>(feat, W0, b0, W, b, out, wh);

    // 3) gather/scale/segment-mean scatter, accumulated onto h0
    scatter_mean_kernel<<<(NETYPE * NEDGES + 7) / 8, 256, 0, stream>>>(
        wh, src, dst, ew, cnt, out);
}